// MixtureGaussianHMMLayer_5995774345471
// MI455X (gfx1250) — compile-verified
//
#include <hip/hip_runtime.h>
#include <hip/hip_bf16.h>

typedef __attribute__((ext_vector_type(2))) float v2f;
typedef __attribute__((ext_vector_type(8))) float v8f;

#define D_FEAT 80
#define N_STATES 32
#define N_COMP 4
#define SC 128            // N_STATES * N_COMP
#define LOG2PI 1.8378770664093453f
#define EPSV 1e-8f

// -------- workspace layout (floats) --------
// Bpack : 40 pairs * 2 halves * 128 sc * 2 vals = 20480   (WMMA-ready B)
// bias  : 128
// Amat  : 32*32 = 1024   (row-softmax of transition logits, clipped at EPS)
// obslp : BT * 32
#define WS_BPACK 0
#define WS_BIAS  20480
#define WS_AMAT  20608
#define WS_OBSLP 21632

// ---------------------------------------------------------------------------
// Kernel 0: pack B matrix / bias / transition softmax.  One block, 128 thr.
// ---------------------------------------------------------------------------
__global__ void hmm_prep_kernel(const float* __restrict__ tlog,
                                const float* __restrict__ wlog,
                                const float* __restrict__ means,
                                const float* __restrict__ lvars,
                                float* __restrict__ Bpack,
                                float* __restrict__ bias,
                                float* __restrict__ Amat) {
    int t = threadIdx.x;                 // sc index 0..127
    int s = t >> 2;
    int c = t & 3;

    // log softmax of mixture weights for (s, c), faithful safe_log(softmax)
    float w0 = wlog[s * 4 + 0], w1 = wlog[s * 4 + 1];
    float w2 = wlog[s * 4 + 2], w3 = wlog[s * 4 + 3];
    float wm = fmaxf(fmaxf(w0, w1), fmaxf(w2, w3));
    float wsum = __expf(w0 - wm) + __expf(w1 - wm) + __expf(w2 - wm) + __expf(w3 - wm);
    float smx = __expf(wlog[s * 4 + c] - wm) / wsum;
    float logw = __logf(fmaxf(smx, EPSV));

    // pack B in WMMA fragment order + accumulate bias terms
    float c1 = 0.f, ld = 0.f;
    for (int d = 0; d < D_FEAT; ++d) {
        float lv = lvars[t * D_FEAT + d];
        float iv = __expf(-lv);
        float mn = means[t * D_FEAT + d];
        int pair = d >> 1, w = d & 1;
        // layout: [pair][half][sc][w]  (half 0: -0.5*iv  half 1: mean*iv)
        Bpack[((pair * 2 + 0) * SC + t) * 2 + w] = -0.5f * iv;
        Bpack[((pair * 2 + 1) * SC + t) * 2 + w] = mn * iv;
        c1 = fmaf(mn * mn, iv, c1);
        ld += lv;
    }
    bias[t] = logw - 0.5f * (c1 + ld + (float)D_FEAT * LOG2PI);

    // transition softmax rows (normal space, clipped at EPS to mirror safe_log)
    if (t < N_STATES) {
        float mx = -INFINITY;
        for (int j = 0; j < N_STATES; ++j) mx = fmaxf(mx, tlog[t * N_STATES + j]);
        float ssum = 0.f;
        for (int j = 0; j < N_STATES; ++j) ssum += __expf(tlog[t * N_STATES + j] - mx);
        for (int j = 0; j < N_STATES; ++j)
            Amat[t * N_STATES + j] = fmaxf(__expf(tlog[t * N_STATES + j] - mx) / ssum, EPSV);
    }
}

// ---------------------------------------------------------------------------
// Kernel 1: fp32 WMMA GEMM  (BT x 160) @ (160 x 128)  ->  -0.5*maha core,
// then + bias and logsumexp over the 4 mixture components (4 adjacent lanes).
// One wave per 16-row M tile; 4 waves per block.
// ---------------------------------------------------------------------------
__global__ void hmm_obslp_wmma_kernel(const float* __restrict__ obs,
                                      const float* __restrict__ Bpack,
                                      const float* __restrict__ bias,
                                      float* __restrict__ obslp,
                                      int BT, int tilesM) {
    const int lane  = threadIdx.x & 31;
    const int wave  = threadIdx.x >> 5;
    const int tileM = blockIdx.x * 4 + wave;
    if (tileM >= tilesM) return;          // wave-uniform: EXEC stays all-ones

    const int m_lane = lane & 15;         // N within tile for B/C, M row for A
    const int khalf  = lane >> 4;         // which K half of the 16x4 A frag

    int rowA = tileM * 16 + m_lane;
    int rA = rowA < BT ? rowA : BT - 1;   // clamp OOB reads
    const float* obsrow = obs + (size_t)rA * D_FEAT;
    const v2f* bbase = (const v2f*)Bpack;

    v8f acc[8] = {};

    #pragma unroll 4
    for (int pair = 0; pair < D_FEAT / 2; ++pair) {
        float x = obsrow[2 * pair + khalf];
        v2f a;
        a.x = x * x;                      // k = 0 (or 2): squared term
        a.y = x;                          // k = 1 (or 3): linear term
        const v2f* brow = bbase + (pair * 2 + khalf) * SC;
        #pragma unroll
        for (int nt = 0; nt < 8; ++nt) {
            v2f b = brow[nt * 16 + m_lane];   // single b64 load per fragment
            acc[nt] = __builtin_amdgcn_wmma_f32_16x16x4_f32(
                false, a, false, b, (short)0, acc[nt], false, false);
        }
    }

    // acc[nt][r] holds -0.5*maha for (bt = tileM*16 + r + 8*khalf, sc = nt*16 + m_lane)
    #pragma unroll
    for (int nt = 0; nt < 8; ++nt) {
        float bv = bias[nt * 16 + m_lane];
        #pragma unroll
        for (int r = 0; r < 8; ++r) {
            int bt = tileM * 16 + r + 8 * khalf;
            float v = acc[nt][r] + bv;
            // logsumexp across the 4 component lanes (xor-1, xor-2 butterfly)
            float mx = fmaxf(v, __shfl_xor(v, 1));
            mx = fmaxf(mx, __shfl_xor(mx, 2));
            float mm = __builtin_isinf(mx) ? 0.f : mx;
            float e = __expf(v - mm);
            float es = e + __shfl_xor(e, 1);
            es = es + __shfl_xor(es, 2);
            float lse = __logf(fmaxf(es, EPSV)) + mm;
            if ((lane & 3) == 0 && bt < BT) {
                int st = nt * 4 + (m_lane >> 2);
                obslp[(size_t)bt * N_STATES + st] = lse;
            }
        }
    }
}

// ---------------------------------------------------------------------------
// Kernel 2: sequential HMM forward recursion in log space.
// One wave per batch; lane j = state j. LSE via normal-space matvec.
// ---------------------------------------------------------------------------
__global__ void hmm_forward_kernel(const float* __restrict__ obslp,
                                   const float* __restrict__ Amat,
                                   float* __restrict__ out,
                                   int B, int T) {
    const int lane = threadIdx.x & 31;
    const int b = threadIdx.x >> 5;
    if (b >= B) return;

    float Areg[N_STATES];
    #pragma unroll
    for (int i = 0; i < N_STATES; ++i) Areg[i] = Amat[i * N_STATES + lane];

    const float* lp = obslp + (size_t)b * T * N_STATES;
    float alpha = lp[lane] - __logf((float)N_STATES);

    for (int t = 1; t < T; ++t) {
        float mx = alpha;
        #pragma unroll
        for (int sh = 1; sh < 32; sh <<= 1) mx = fmaxf(mx, __shfl_xor(mx, sh));
        float mm = __builtin_isinf(mx) ? 0.f : mx;
        float p = __expf(alpha - mm);
        float s = 0.f;
        #pragma unroll
        for (int i = 0; i < N_STATES; ++i)
            s = fmaf(__shfl(p, i), Areg[i], s);   // readlane broadcast matvec
        alpha = lp[(size_t)t * N_STATES + lane] + __logf(fmaxf(s, EPSV)) + mm;
    }

    // final logsumexp over states
    float mx = alpha;
    #pragma unroll
    for (int sh = 1; sh < 32; sh <<= 1) mx = fmaxf(mx, __shfl_xor(mx, sh));
    float mm = __builtin_isinf(mx) ? 0.f : mx;
    float e = __expf(alpha - mm);
    #pragma unroll
    for (int sh = 1; sh < 32; sh <<= 1) e += __shfl_xor(e, sh);
    if (lane == 0) out[b] = __logf(fmaxf(e, EPSV)) + mm;
}

// ---------------------------------------------------------------------------
extern "C" void kernel_launch(void* const* d_in, const int* in_sizes, int n_in,
                              void* d_out, int out_size, void* d_ws, size_t ws_size,
                              hipStream_t stream) {
    const float* obs   = (const float*)d_in[0];   // (B, T, 80)
    const float* tlog  = (const float*)d_in[1];   // (32, 32)
    const float* wlog  = (const float*)d_in[2];   // (32, 4)
    const float* means = (const float*)d_in[3];   // (32, 4, 80)
    const float* lvars = (const float*)d_in[4];   // (32, 4, 80)
    float* out = (float*)d_out;

    int B  = out_size;                     // 8
    int BT = in_sizes[0] / D_FEAT;         // 8000
    int T  = BT / B;                       // 1000

    float* ws     = (float*)d_ws;
    float* Bpack  = ws + WS_BPACK;
    float* bias   = ws + WS_BIAS;
    float* Amat   = ws + WS_AMAT;
    float* obslp  = ws + WS_OBSLP;

    hmm_prep_kernel<<<1, 128, 0, stream>>>(tlog, wlog, means, lvars, Bpack, bias, Amat);

    int tilesM = (BT + 15) / 16;
    int blocks = (tilesM + 3) / 4;
    hmm_obslp_wmma_kernel<<<blocks, 128, 0, stream>>>(obs, Bpack, bias, obslp, BT, tilesM);

    hmm_forward_kernel<<<1, 32 * B, 0, stream>>>(obslp, Amat, out, B, T);
}